// MaxSW_38242388804119
// MI455X (gfx1250) — compile-verified
//
#include <hip/hip_runtime.h>
#include <hip/hip_bf16.h>

// MaxSW (max sliced Wasserstein) for B=128, N=32768, D=3, P=1.
// One persistent workgroup per batch; both projection arrays live in LDS
// (256KB, possible only on CDNA5's 320KB/WGP LDS); bitonic sort in LDS;
// gradient via binary-search ranks; projection via V_WMMA_F32_16X16X4_F32.

#define MSW_N        32768
#define MSW_B        128
#define MSW_THREADS  1024
#define MSW_WAVES    (MSW_THREADS / 32)
#define MSW_LR       0.01f

typedef __attribute__((ext_vector_type(2))) float v2f;
typedef __attribute__((ext_vector_type(8))) float v8f;

#if defined(__gfx1250__) && __has_builtin(__builtin_amdgcn_wmma_f32_16x16x4_f32)
#define MSW_USE_WMMA 1
#else
#define MSW_USE_WMMA 0
#endif

// Project N points (row-major [N][3]) onto (p0,p1,p2), results -> LDS dst[N].
// WMMA path: A(16x4) = proj replicated in every row, B(4x16) = 16 points as
// columns, D(16x16) row 0 = 16 projections. Per 32-bit fragment layouts:
//   A: lanes 0-15 hold K={0,1} in v[0],v[1]; lanes 16-31 hold K={2,3}.
//   B: lanes 0-15 hold rows K=0/K=1 (col = lane); lanes 16-31 rows K=2/K=3.
//   D: VGPR0, lanes 0-15 = row M=0, cols N=0..15.
__device__ __forceinline__ void msw_project(const float* __restrict__ src,
                                            float* __restrict__ dst,
                                            float p0, float p1, float p2,
                                            unsigned tid)
{
#if MSW_USE_WMMA
    const unsigned lane = tid & 31u;
    const unsigned wave = tid >> 5u;
    const bool     hi   = (lane >= 16u);
    const unsigned col  = lane & 15u;

    v2f a;
    a.x = hi ? p2 : p0;
    a.y = hi ? 0.0f : p1;   // K=3 pad / K=1

    for (unsigned n0 = wave * 16u; n0 < MSW_N; n0 += MSW_WAVES * 16u) {
        const unsigned idx = (n0 + col) * 3u;
        const float e0 = src[idx + 0];
        const float e1 = src[idx + 1];
        const float e2 = src[idx + 2];
        v2f b;
        b.x = hi ? e2 : e0;
        b.y = hi ? 0.0f : e1;
        v8f acc = {};
        acc = __builtin_amdgcn_wmma_f32_16x16x4_f32(
            /*neg_a=*/false, a, /*neg_b=*/false, b,
            /*c_mod=*/(short)0, acc, /*reuse_a=*/false, /*reuse_b=*/false);
        if (!hi) dst[n0 + col] = acc[0];   // row M=0 of D
    }
#else
    for (unsigned n = tid; n < MSW_N; n += MSW_THREADS) {
        const unsigned idx = n * 3u;
        dst[n] = fmaf(src[idx], p0, fmaf(src[idx + 1], p1, src[idx + 2] * p2));
    }
#endif
}

// Sort two independent N-element segments (data[0..N), data[N..2N)) ascending
// with one fused bitonic network. j <= N/2 < N so partners never cross the
// segment boundary; direction uses the segment-local index.
__device__ __forceinline__ void msw_bitonic_sort_pair(float* data, unsigned tid)
{
    for (unsigned k = 2u; k <= MSW_N; k <<= 1) {
        for (unsigned j = k >> 1; j > 0u; j >>= 1) {
            for (unsigned i = tid; i < 2u * MSW_N; i += MSW_THREADS) {
                const unsigned ixj = i ^ j;
                if (ixj > i) {
                    const float va = data[i];
                    const float vb = data[ixj];
                    const bool up = (((i & (MSW_N - 1u)) & k) == 0u);
                    if ((va > vb) == up) { data[i] = vb; data[ixj] = va; }
                }
            }
            __syncthreads();
        }
    }
}

// rank = #elements strictly less than key (15 or 16 iterations, key-dependent).
__device__ __forceinline__ int msw_lower_bound(const float* a, float key)
{
    int lo = 0, hi = MSW_N;
    while (lo < hi) {
        const int mid = (lo + hi) >> 1;
        if (a[mid] < key) lo = mid + 1; else hi = mid;
    }
    return (lo < MSW_N) ? lo : (MSW_N - 1);
}

__device__ __forceinline__ float msw_wave_sum(float v)
{
    for (int off = 16; off > 0; off >>= 1) v += __shfl_down(v, off, 32);
    return v;
}

__global__ __launch_bounds__(MSW_THREADS)
void maxsw_main_kernel(const float* __restrict__ x,
                       const float* __restrict__ y,
                       const float* __restrict__ proj_init,
                       const int*   __restrict__ num_iters_p,
                       float*       __restrict__ out,      // [0]=loss, [1..3B]=proj
                       float*       __restrict__ moments)  // d_ws: B floats
{
    extern __shared__ float lds[];
    float* xs  = lds;                       // N floats (xp, then sorted)
    float* ys  = lds + MSW_N;               // N floats (yp, then sorted)
    float* red = lds + 2u * MSW_N;          // MSW_WAVES * 8 scratch
    float* prj = red + MSW_WAVES * 8u;      // 3 floats

    const unsigned tid = threadIdx.x;
    const unsigned b   = blockIdx.x;
    const float* xb = x + (size_t)b * MSW_N * 3u;
    const float* yb = y + (size_t)b * MSW_N * 3u;
    const int iters = *num_iters_p;

    if (tid < 3u) prj[tid] = proj_init[b * 3u + tid];
    __syncthreads();

    for (int it = 0; it <= iters; ++it) {
        const float p0 = prj[0], p1 = prj[1], p2 = prj[2];

        // 1) projections -> LDS (WMMA)
        msw_project(xb, xs, p0, p1, p2, tid);
        msw_project(yb, ys, p0, p1, p2, tid);
        __syncthreads();

        // 2) sort both arrays ascending in LDS
        msw_bitonic_sort_pair(lds, tid);

        // 3) s = sum (xs-ys)^2 ; gradient terms via binary-search ranks
        const bool need_grad = (it < iters);
        float s = 0.0f;
        float gx0 = 0.0f, gx1 = 0.0f, gx2 = 0.0f;
        float gy0 = 0.0f, gy1 = 0.0f, gy2 = 0.0f;
        for (unsigned i = tid; i < MSW_N; i += MSW_THREADS) {
            const float d = xs[i] - ys[i];
            s = fmaf(d, d, s);
            if (need_grad) {
                const unsigned idx = i * 3u;
                __builtin_prefetch(xb + idx + 3u * MSW_THREADS, 0, 1);
                __builtin_prefetch(yb + idx + 3u * MSW_THREADS, 0, 1);
                const float xe0 = xb[idx], xe1 = xb[idx + 1], xe2 = xb[idx + 2];
                const float px = fmaf(xe0, p0, fmaf(xe1, p1, xe2 * p2));
                const int r = msw_lower_bound(xs, px);
                const float w = px - ys[r];          // diff at x's sorted slot
                gx0 = fmaf(w, xe0, gx0);
                gx1 = fmaf(w, xe1, gx1);
                gx2 = fmaf(w, xe2, gx2);
                const float ye0 = yb[idx], ye1 = yb[idx + 1], ye2 = yb[idx + 2];
                const float py = fmaf(ye0, p0, fmaf(ye1, p1, ye2 * p2));
                const int r2 = msw_lower_bound(ys, py);
                const float w2 = xs[r2] - py;        // diff at y's sorted slot
                gy0 = fmaf(w2, ye0, gy0);
                gy1 = fmaf(w2, ye1, gy1);
                gy2 = fmaf(w2, ye2, gy2);
            }
        }

        // 4) block reduction (wave shuffle, then thread 0)
        s   = msw_wave_sum(s);
        gx0 = msw_wave_sum(gx0); gx1 = msw_wave_sum(gx1); gx2 = msw_wave_sum(gx2);
        gy0 = msw_wave_sum(gy0); gy1 = msw_wave_sum(gy1); gy2 = msw_wave_sum(gy2);
        const unsigned lane = tid & 31u, wv = tid >> 5u;
        if (lane == 0u) {
            float* r8 = red + wv * 8u;
            r8[0] = s;
            r8[1] = gx0; r8[2] = gx1; r8[3] = gx2;
            r8[4] = gy0; r8[5] = gy1; r8[6] = gy2;
        }
        __syncthreads();

        if (tid == 0u) {
            float S = 0.f, G0 = 0.f, G1 = 0.f, G2 = 0.f, H0 = 0.f, H1 = 0.f, H2 = 0.f;
            for (unsigned wi = 0; wi < MSW_WAVES; ++wi) {
                const float* r8 = red + wi * 8u;
                S  += r8[0];
                G0 += r8[1]; G1 += r8[2]; G2 += r8[3];
                H0 += r8[4]; H1 += r8[5]; H2 += r8[6];
            }
            const float rs = sqrtf(S);
            if (it < iters) {
                // L = mean_b(-sqrt(s_b)) => dL/dproj = -(1/(B*sqrt(s)))*(gx-gy)
                const float c = -1.0f / ((float)MSW_B * fmaxf(rs, 1e-30f));
                const float g0 = c * (G0 - H0);
                const float g1 = c * (G1 - H1);
                const float g2 = c * (G2 - H2);
                const float n0 = p0 - MSW_LR * g0;
                const float n1 = p1 - MSW_LR * g1;
                const float n2 = p2 - MSW_LR * g2;
                const float inv = rsqrtf(fmaf(n0, n0, fmaf(n1, n1, n2 * n2)));
                prj[0] = n0 * inv; prj[1] = n1 * inv; prj[2] = n2 * inv;
            } else {
                moments[b] = rs;                  // first_moment[b] = sqrt(s)
                out[1 + b * 3u + 0] = p0;         // final proj (already normalized)
                out[1 + b * 3u + 1] = p1;
                out[1 + b * 3u + 2] = p2;
            }
        }
        __syncthreads();
    }
}

__global__ __launch_bounds__(MSW_B)
void maxsw_reduce_kernel(const float* __restrict__ moments, float* __restrict__ out)
{
    __shared__ float sh[MSW_B];
    const unsigned t = threadIdx.x;
    sh[t] = moments[t];
    __syncthreads();
    for (unsigned off = MSW_B / 2u; off > 0u; off >>= 1) {
        if (t < off) sh[t] += sh[t + off];
        __syncthreads();
    }
    if (t == 0u) out[0] = sh[0] / (float)MSW_B;   // loss = mean_b sqrt(s_b)
}

extern "C" void kernel_launch(void* const* d_in, const int* in_sizes, int n_in,
                              void* d_out, int out_size, void* d_ws, size_t ws_size,
                              hipStream_t stream)
{
    const float* x         = (const float*)d_in[0];
    const float* y         = (const float*)d_in[1];
    const float* proj_init = (const float*)d_in[2];
    const int*   num_iters = (const int*)d_in[3];
    float* out     = (float*)d_out;
    float* moments = (float*)d_ws;   // MSW_B floats of scratch

    const int B = in_sizes[2] / 3;   // 128

    const size_t shmem = (size_t)(2u * MSW_N + MSW_WAVES * 8u + 8u) * sizeof(float);
    (void)hipFuncSetAttribute((const void*)maxsw_main_kernel,
                              hipFuncAttributeMaxDynamicSharedMemorySize,
                              (int)shmem);

    maxsw_main_kernel<<<dim3(B), dim3(MSW_THREADS), shmem, stream>>>(
        x, y, proj_init, num_iters, out, moments);
    maxsw_reduce_kernel<<<dim3(1), dim3(MSW_B), 0, stream>>>(moments, out);
}